// ComplexBatchNorm2d_14757507629807
// MI455X (gfx1250) — compile-verified
//
#include <hip/hip_runtime.h>
#include <hip/hip_bf16.h>

typedef __attribute__((ext_vector_type(2))) float v2f;
typedef __attribute__((ext_vector_type(8))) float v8f;

#define NB 32
#define NC 64
#define HW 16384          // 128*128
#define NTOT 524288.0f    // NB*HW
#define CBN_EPS 1e-5f

// ws layout (floats): P[NC][NB][8] partials (q: Sr,Si,Srr,Sii,Sri,pad..), then params[NC][8]
#define P_STRIDE 8
#define PARAM_OFF (NC * NB * P_STRIDE)

__device__ __forceinline__ float wave_reduce_add(float v) {
#pragma unroll
    for (int m = 16; m > 0; m >>= 1) v += __shfl_xor(v, m, 32);
    return v;
}

// ---------------- Stage 1: per-(b,c) slab partial sums --------------------
__global__ void cbn_reduce_kernel(const float* __restrict__ xr,
                                  const float* __restrict__ xi,
                                  float* __restrict__ P) {
    const int b = blockIdx.x;          // slab = batch index
    const int c = blockIdx.y;
    const int tid = threadIdx.x;       // 256 threads = 8 waves
    const size_t base = ((size_t)b * NC + c) * HW;
    const float4* r4 = (const float4*)(xr + base);
    const float4* i4 = (const float4*)(xi + base);

    float sr = 0.f, si = 0.f, srr = 0.f, sii = 0.f, sri = 0.f;
#pragma unroll
    for (int it = 0; it < 16; ++it) {   // 16 * 256 float4 = 16384 floats
        const int idx = it * 256 + tid;
        if (it < 12) {  // speculative prefetch a few KB ahead
            __builtin_prefetch((const void*)(r4 + idx + 1024), 0, 0);
            __builtin_prefetch((const void*)(i4 + idx + 1024), 0, 0);
        }
        float4 a = r4[idx];
        float4 q = i4[idx];
        sr  += (a.x + a.y) + (a.z + a.w);
        si  += (q.x + q.y) + (q.z + q.w);
        srr += a.x * a.x + a.y * a.y + a.z * a.z + a.w * a.w;
        sii += q.x * q.x + q.y * q.y + q.z * q.z + q.w * q.w;
        sri += a.x * q.x + a.y * q.y + a.z * q.z + a.w * q.w;
    }

    sr = wave_reduce_add(sr);
    si = wave_reduce_add(si);
    srr = wave_reduce_add(srr);
    sii = wave_reduce_add(sii);
    sri = wave_reduce_add(sri);

    __shared__ float part[8][5];
    const int wave = tid >> 5, lane = tid & 31;
    if (lane == 0) {
        part[wave][0] = sr;  part[wave][1] = si;
        part[wave][2] = srr; part[wave][3] = sii; part[wave][4] = sri;
    }
    __syncthreads();
    if (tid < 5) {  // deterministic cross-wave combine, one stat per thread
        float acc = 0.f;
#pragma unroll
        for (int w = 0; w < 8; ++w) acc += part[w][tid];
        P[((size_t)c * NB + b) * P_STRIDE + tid] = acc;
    }
}

// ------- Stage 2: WMMA cross-slab sum + analytic V^{-1/2}, fold gamma/beta -
// One block, 128 threads = 4 waves; wave w owns channels [16w, 16w+16).
// D = A(ones 16x4) x B(4x16) accumulated over 8 chunks sums 32 slab partials
// per channel (columns) entirely in f32 on the matrix unit.
__global__ void cbn_finalize_kernel(const float* __restrict__ P,
                                    const float* __restrict__ gamma,
                                    const float* __restrict__ beta,
                                    float* __restrict__ params) {
    const int tid = threadIdx.x;
    const int wave = tid >> 5, lane = tid & 31;
    const int c_base = wave * 16;
    const int n = lane & 15;                 // column = channel within group
    const int k0 = (lane < 16) ? 0 : 2;      // B rows held by this half-wave

    v2f a; a.x = 1.0f; a.y = 1.0f;           // all-ones A: layout-invariant

    float tot[5];
#pragma unroll
    for (int q = 0; q < 5; ++q) {
        v8f acc = {};
#pragma unroll
        for (int t = 0; t < 8; ++t) {        // 8 chunks of K=4 slabs
            const float* bp =
                P + ((size_t)(c_base + n) * NB + t * 4 + k0) * P_STRIDE + q;
            v2f bf; bf.x = bp[0]; bf.y = bp[P_STRIDE];
            acc = __builtin_amdgcn_wmma_f32_16x16x4_f32(
                false, a, false, bf, (short)0, acc, false, false);
        }
        tot[q] = acc[0];                     // lanes 0..15: D[0][n] = colsum
    }

    if (lane < 16) {
        const int c = c_base + n;
        const float mr = tot[0] / NTOT, mi = tot[1] / NTOT;
        const float inv_nm1 = 1.0f / (NTOT - 1.0f);
        const float Vrr = (tot[2] - NTOT * mr * mr) * inv_nm1;
        const float Vii = (tot[3] - NTOT * mi * mi) * inv_nm1;
        const float Vri = (tot[4] - NTOT * mr * mi) * inv_nm1;
        // (V + eps I)^{-1/2} via Cayley-Hamilton
        const float A = Vrr + CBN_EPS, D = Vii + CBN_EPS, Bv = Vri;
        const float s = sqrtf(A * D - Bv * Bv);
        const float t = sqrtf(A + D + 2.0f * s);
        const float inv = 1.0f / (s * t);
        const float W00 = (D + s) * inv, W01 = -Bv * inv, W11 = (A + s) * inv;
        const float g00 = gamma[c * 4 + 0], g01 = gamma[c * 4 + 1];
        const float g10 = gamma[c * 4 + 2], g11 = gamma[c * 4 + 3];
        const float M00 = g00 * W00 + g01 * W01;   // W10 == W01
        const float M01 = g00 * W01 + g01 * W11;
        const float M10 = g10 * W00 + g11 * W01;
        const float M11 = g10 * W01 + g11 * W11;
        const float b0 = beta[c * 2 + 0] - (M00 * mr + M01 * mi);
        const float b1 = beta[c * 2 + 1] - (M10 * mr + M11 * mi);
        float* p = params + c * 8;
        p[0] = M00; p[1] = M01; p[2] = M10; p[3] = M11; p[4] = b0; p[5] = b1;
    }
}

// ---------------- Stage 3: streaming apply y = M x + b --------------------
__global__ void cbn_apply_kernel(const float* __restrict__ xr,
                                 const float* __restrict__ xi,
                                 const float* __restrict__ params,
                                 float* __restrict__ out) {
    const int c = blockIdx.y, b = blockIdx.z;
    const float* p = params + c * 8;
    const float M00 = p[0], M01 = p[1], M10 = p[2], M11 = p[3];
    const float b0 = p[4], b1 = p[5];

    const size_t base = ((size_t)b * NC + c) * HW;
    const int idx4 = blockIdx.x * 256 + threadIdx.x;   // float4 index
    const float4* r4 = (const float4*)(xr + base);
    const float4* i4 = (const float4*)(xi + base);
    float4 a = r4[idx4];
    float4 q = i4[idx4];

    float4 o0, o1;
    o0.x = M00 * a.x + M01 * q.x + b0;  o0.y = M10 * a.x + M11 * q.x + b1;
    o0.z = M00 * a.y + M01 * q.y + b0;  o0.w = M10 * a.y + M11 * q.y + b1;
    o1.x = M00 * a.z + M01 * q.z + b0;  o1.y = M10 * a.z + M11 * q.z + b1;
    o1.z = M00 * a.w + M01 * q.w + b0;  o1.w = M10 * a.w + M11 * q.w + b1;

    float4* o = (float4*)(out + base * 2);             // (..., H, W, 2) interleaved
    o[idx4 * 2 + 0] = o0;
    o[idx4 * 2 + 1] = o1;
}

extern "C" void kernel_launch(void* const* d_in, const int* in_sizes, int n_in,
                              void* d_out, int out_size, void* d_ws, size_t ws_size,
                              hipStream_t stream) {
    const float* xr    = (const float*)d_in[0];
    const float* xi    = (const float*)d_in[1];
    const float* gamma = (const float*)d_in[2];
    const float* beta  = (const float*)d_in[3];
    float* out = (float*)d_out;
    float* P = (float*)d_ws;                 // NC*NB*8 floats (all used slots written)
    float* params = P + PARAM_OFF;           // NC*8 floats

    cbn_reduce_kernel<<<dim3(NB, NC), 256, 0, stream>>>(xr, xi, P);
    cbn_finalize_kernel<<<1, 128, 0, stream>>>(P, gamma, beta, params);
    cbn_apply_kernel<<<dim3(HW / 1024, NC, NB), 256, 0, stream>>>(xr, xi, params, out);
}